// GIN_67018669687296
// MI455X (gfx1250) — compile-verified
//
#include <hip/hip_runtime.h>

#define D      128
#define HS     132          // padded LDS row stride (bank-conflict-free b64 frag loads)
#define BN_EPS 1e-5f

typedef float v2f __attribute__((ext_vector_type(2)));
typedef float v8f __attribute__((ext_vector_type(8)));
typedef unsigned int v4u __attribute__((ext_vector_type(4)));
typedef int v8i __attribute__((ext_vector_type(8)));
typedef int v4i __attribute__((ext_vector_type(4)));

#if __has_builtin(__builtin_amdgcn_tensor_store_from_lds) && __has_builtin(__builtin_amdgcn_s_wait_tensorcnt)
#define HAVE_TDM 1
#else
#define HAVE_TDM 0
#endif

// ---------------------------------------------------------------- zero fill
__global__ void gin_zero_f4(float4* __restrict__ p, int n4) {
    int i = blockIdx.x * blockDim.x + threadIdx.x;
    int stride = gridDim.x * blockDim.x;
    float4 z = make_float4(0.f, 0.f, 0.f, 0.f);
    for (; i < n4; i += stride) p[i] = z;
}

__global__ void gin_zero_stats(float* __restrict__ p, int n) {
    int i = blockIdx.x * blockDim.x + threadIdx.x;
    if (i < n) p[i] = 0.f;
}

// ---------------------------------------------------- edge scatter (segment_sum)
__global__ void gin_scatter_add(const float* __restrict__ xin,
                                const int* __restrict__ esrc,
                                const int* __restrict__ edst,
                                float* __restrict__ agg, int nEdges) {
    int idx = blockIdx.x * blockDim.x + threadIdx.x;
    int e = idx >> 5;
    if (e >= nEdges) return;
    int c4 = (idx & 31) << 2;
    int s = esrc[e];
    int d = edst[e];
    const float4 v = *(const float4*)&xin[s * D + c4];
    float* dst = &agg[d * D + c4];
    unsafeAtomicAdd(dst + 0, v.x);
    unsafeAtomicAdd(dst + 1, v.y);
    unsafeAtomicAdd(dst + 2, v.z);
    unsafeAtomicAdd(dst + 3, v.w);
}

// -------------------------------------------- fused (x+agg) -> MLP -> ReLU -> BN stats
// 256 threads = 8 waves; block handles 128 node rows; each wave owns 16 output rows.
__global__ __launch_bounds__(256) void gin_mlp(
    const float* __restrict__ xin, const float* __restrict__ agg,
    const float* __restrict__ W1, const float* __restrict__ b1,
    const float* __restrict__ W2, const float* __restrict__ b2,
    float* __restrict__ Y, float* __restrict__ gstat, int nNodes)
{
    __shared__ float sH[D * HS];      // H tile (padded rows); reused in-place for T
    __shared__ float sWt[D * HS];     // W transposed: sWt[col*HS + k]; reused as Y staging
    __shared__ float sStat[2 * D];    // per-block column sum / sumsq

    const int tid  = threadIdx.x;
    const int r0   = blockIdx.x * D;
    const int wave = tid >> 5;
    const int lane = tid & 31;
    const int ln   = lane & 15;       // M (A) / N (B,C,D) index within 16
    const int hi   = lane >> 4;
    const int kk   = hi * 2;          // lanes 16-31 hold K=2,3
    const int m0   = wave * 16;

    // ---- stage H = x + agg (zero-pad OOB rows) ----
    for (int i = tid; i < D * (D / 4); i += 256) {
        int row  = i >> 5;
        int c4   = (i & 31) << 2;
        int grow = r0 + row;
        float4 v;
        if (grow < nNodes) {
            float4 a = *(const float4*)&xin[grow * D + c4];
            float4 g = *(const float4*)&agg[grow * D + c4];
            v = make_float4(a.x + g.x, a.y + g.y, a.z + g.z, a.w + g.w);
        } else {
            v = make_float4(0.f, 0.f, 0.f, 0.f);
        }
        *(float4*)&sH[row * HS + c4] = v;
    }
    // ---- stage W1 transposed: sWt[col][k] = W1[k][col] ----
    for (int i = tid; i < (D * D) / 4; i += 256) {
        float4 w = *(const float4*)&W1[i * 4];
        int row = i >> 5;            // k
        int col = (i & 31) << 2;
        sWt[(col + 0) * HS + row] = w.x;
        sWt[(col + 1) * HS + row] = w.y;
        sWt[(col + 2) * HS + row] = w.z;
        sWt[(col + 3) * HS + row] = w.w;
    }
    sStat[tid] = 0.f;
    __syncthreads();

    const v8f zero8 = {0.f, 0.f, 0.f, 0.f, 0.f, 0.f, 0.f, 0.f};
    v8f acc[8];
    const float* aRow = &sH[(m0 + ln) * HS + kk];    // A frag base: this wave's rows
    const float* bCol = &sWt[ln * HS + kk];          // B frag base: per-lane column

    // ================= GEMM1: T = ReLU(H @ W1 + b1) =================
    #pragma unroll
    for (int nt = 0; nt < 8; ++nt) acc[nt] = zero8;

    for (int k = 0; k < D; k += 4) {
        v2f a = *(const v2f*)&aRow[k];               // ds_load_b64, aligned pair
        #pragma unroll
        for (int nt = 0; nt < 8; ++nt) {
            v2f b = *(const v2f*)&bCol[nt * 16 * HS + k];   // ds_load_b64, aligned pair
            acc[nt] = __builtin_amdgcn_wmma_f32_16x16x4_f32(
                false, a, false, b, (short)0, acc[nt], false, false);
        }
    }
    // epilogue 1: bias + ReLU, write T into this wave's own sH rows
    #pragma unroll
    for (int nt = 0; nt < 8; ++nt) {
        int col = nt * 16 + ln;
        float bb = b1[col];
        #pragma unroll
        for (int j = 0; j < 8; ++j) {
            float t = fmaxf(acc[nt][j] + bb, 0.f);
            sH[(m0 + j + 8 * hi) * HS + col] = t;
        }
    }
    __syncthreads();
    // ---- stage W2 transposed ----
    for (int i = tid; i < (D * D) / 4; i += 256) {
        float4 w = *(const float4*)&W2[i * 4];
        int row = i >> 5;
        int col = (i & 31) << 2;
        sWt[(col + 0) * HS + row] = w.x;
        sWt[(col + 1) * HS + row] = w.y;
        sWt[(col + 2) * HS + row] = w.z;
        sWt[(col + 3) * HS + row] = w.w;
    }
    __syncthreads();

    // ================= GEMM2: Yt = T @ W2 + b2 =================
    #pragma unroll
    for (int nt = 0; nt < 8; ++nt) acc[nt] = zero8;

    for (int k = 0; k < D; k += 4) {
        v2f a = *(const v2f*)&aRow[k];
        #pragma unroll
        for (int nt = 0; nt < 8; ++nt) {
            v2f b = *(const v2f*)&bCol[nt * 16 * HS + k];
            acc[nt] = __builtin_amdgcn_wmma_f32_16x16x4_f32(
                false, a, false, b, (short)0, acc[nt], false, false);
        }
    }

#if HAVE_TDM
    __syncthreads();                  // all waves done reading sWt (W2): safe to reuse as sY
    float* sY = sWt;                  // contiguous 128x128 staging for TDM store
    #pragma unroll
    for (int nt = 0; nt < 8; ++nt) {
        int col = nt * 16 + ln;
        float bb = b2[col];
        float s = 0.f, q = 0.f;
        #pragma unroll
        for (int j = 0; j < 8; ++j) {
            float y = fmaxf(acc[nt][j] + bb, 0.f);
            int row = m0 + j + 8 * hi;
            sY[row * D + col] = y;                   // conflict-free (consecutive banks)
            if (r0 + row < nNodes) { s += y; q += y * y; }
        }
        atomicAdd(&sStat[col], s);
        atomicAdd(&sStat[D + col], q);
    }
    __syncthreads();                  // Y tile staged
    if (wave == 0) {
        // ---- Tensor DMA descriptor: store 128x(N-r0 clamped) f32 tile LDS -> Y + r0*D ----
        unsigned long long ga = (unsigned long long)(uintptr_t)(Y + (long)r0 * D);
        unsigned rows = (unsigned)(nNodes - r0);     // TDM drops writes beyond tensor_dim1
        v4u g0;
        g0[0] = 1u;                                               // count=1, user mode
        g0[1] = (unsigned)(uintptr_t)(void*)sY;                   // lds_addr
        g0[2] = (unsigned)(ga & 0xFFFFFFFFu);                     // global_addr[31:0]
        g0[3] = (unsigned)((ga >> 32) & 0x01FFFFFFu) | (2u << 30); // global_addr[56:32] | type=2
        v8i g1;
        g1[0] = (int)(2u << 16);                                  // data_size = 4B
        g1[1] = (int)(128u << 16);                                // tensor_dim0 = 128 (lo16)
        g1[2] = (int)((rows & 0xFFFFu) << 16);                    // dim0 hi=0 | tensor_dim1 lo16
        g1[3] = (int)(((rows >> 16) & 0xFFFFu) | (128u << 16));   // tensor_dim1 hi | tile_dim0=128
        g1[4] = 128;                                              // tile_dim1 = 128
        g1[5] = 128;                                              // tensor_dim0_stride = 128
        g1[6] = 0;
        g1[7] = 0;
        v4i gz4 = {0, 0, 0, 0};
        v8i gz8 = {0, 0, 0, 0, 0, 0, 0, 0};
        __builtin_amdgcn_tensor_store_from_lds(g0, g1, gz4, gz4, gz8, 0);
        __builtin_amdgcn_s_wait_tensorcnt(0);
    }
#else
    #pragma unroll
    for (int nt = 0; nt < 8; ++nt) {
        int col = nt * 16 + ln;
        float bb = b2[col];
        float s = 0.f, q = 0.f;
        #pragma unroll
        for (int j = 0; j < 8; ++j) {
            float y = fmaxf(acc[nt][j] + bb, 0.f);
            int grow = r0 + m0 + j + 8 * hi;
            if (grow < nNodes) {
                Y[grow * D + col] = y;
                s += y;
                q += y * y;
            }
        }
        atomicAdd(&sStat[col], s);
        atomicAdd(&sStat[D + col], q);
    }
    __syncthreads();
#endif
    if (tid < 2 * D) unsafeAtomicAdd(&gstat[tid], sStat[tid]);
}

// ------------------------------------------------ BN: finalize scale/shift
__global__ void gin_bn_finalize(const float* __restrict__ gstat,
                                const float* __restrict__ gamma,
                                const float* __restrict__ beta,
                                float* __restrict__ ss, float invN) {
    int c = threadIdx.x;
    if (c < D) {
        float mean = gstat[c] * invN;
        float var  = gstat[D + c] * invN - mean * mean;
        float sc   = gamma[c] * rsqrtf(var + BN_EPS);
        ss[c]      = sc;
        ss[D + c]  = beta[c] - mean * sc;
    }
}

// ------------------------------------------------ BN: apply (vectorized)
__global__ void gin_bn_apply(const float* __restrict__ Y, const float* __restrict__ ss,
                             float* __restrict__ xout, int n4) {
    int i = blockIdx.x * blockDim.x + threadIdx.x;
    int stride = gridDim.x * blockDim.x;
    for (; i < n4; i += stride) {
        int c4 = (i & 31) << 2;
        float4 y = *(const float4*)&Y[(size_t)i * 4];
        float4 o;
        o.x = y.x * ss[c4 + 0] + ss[D + c4 + 0];
        o.y = y.y * ss[c4 + 1] + ss[D + c4 + 1];
        o.z = y.z * ss[c4 + 2] + ss[D + c4 + 2];
        o.w = y.w * ss[c4 + 3] + ss[D + c4 + 3];
        *(float4*)&xout[(size_t)i * 4] = o;
    }
}

// ================================================================ host side
extern "C" void kernel_launch(void* const* d_in, const int* in_sizes, int n_in,
                              void* d_out, int out_size, void* d_ws, size_t ws_size,
                              hipStream_t stream) {
    const float* x     = (const float*)d_in[0];
    const int*   esrc  = (const int*)  d_in[1];
    const int*   edst  = (const int*)  d_in[2];
    const float* W1    = (const float*)d_in[3];
    const float* b1    = (const float*)d_in[4];
    const float* W2    = (const float*)d_in[5];
    const float* b2    = (const float*)d_in[6];
    const float* gamma = (const float*)d_in[7];
    const float* beta  = (const float*)d_in[8];

    const int N  = in_sizes[0] / D;
    const int nE = in_sizes[1];
    const int L  = in_sizes[3] / (D * D);

    const long ND = (long)N * D;
    float* ws    = (float*)d_ws;
    float* agg   = ws;                       // N*D
    float* Ybuf  = ws + ND;                  // N*D
    float* xcur  = ws + 2 * ND;              // N*D
    float* stats = ws + 3 * ND;              // gsum[128], gsumsq[128], scale[128], shift[128]

    const int n4    = (int)(ND / 4);
    const int zgrid = (n4 + 255) / 256;
    const int sgrid = (int)(((long)nE * 32 + 255) / 256);
    const int ggrid = (N + D - 1) / D;
    const float invN = 1.0f / (float)N;
    float* out = (float*)d_out;

    for (int l = 0; l < L; ++l) {
        const float* xin  = (l == 0)     ? x   : xcur;
        float*       xout = (l == L - 1) ? out : xcur;

        gin_zero_f4    <<<zgrid, 256, 0, stream>>>((float4*)agg, n4);
        gin_zero_stats <<<1,     256, 0, stream>>>(stats, 2 * D);
        gin_scatter_add<<<sgrid, 256, 0, stream>>>(xin, esrc, edst, agg, nE);
        gin_mlp        <<<ggrid, 256, 0, stream>>>(xin, agg,
                                                   W1 + (long)l * D * D, b1 + l * D,
                                                   W2 + (long)l * D * D, b2 + l * D,
                                                   Ybuf, stats, N);
        gin_bn_finalize<<<1, D, 0, stream>>>(stats, gamma + l * D, beta + l * D,
                                             stats + 2 * D, invN);
        gin_bn_apply   <<<2048, 256, 0, stream>>>(Ybuf, stats + 2 * D, xout, n4);
    }
}